// HGCNN_3513283248953
// MI455X (gfx1250) — compile-verified
//
#include <hip/hip_runtime.h>
#include <stdint.h>

// ---------------------------------------------------------------------------
// HGCNN forward for MI455X (gfx1250, wave32).
// All GEMMs use V_WMMA_F32_16X16X4_F32 (exact f32 — kNN index selection is
// precision-sensitive). Edge conv uses the gather/1x1-conv commutation:
//   Y[b,n,k,o] = U[b, idx[b,n,k], o] + U[b, n, Co+o],
//   U = X · [wl ; wr - wl]^T
// All reductions are chunked + tree-reduced (no float atomics) => deterministic.
// GEMM: one wave computes a 16x64 tile (4 WMMAs / K-step). OOB rows read a
// zero-filled scratch row (no masking in the loop). The K-loop is ping-pong
// double-buffered (two fragment sets, no rotate movs) so each buffer's loads
// overlap the other buffer's WMMAs with staggered s_wait_loadcnt.
// ---------------------------------------------------------------------------

typedef float v2f __attribute__((ext_vector_type(2)));
typedef float v8f __attribute__((ext_vector_type(8)));

#define BN_EPS 1e-5f
#define SLOPE 0.2f
#define LRELU(v) ((v) > 0.f ? (v) : SLOPE * (v))
#define ZLEN 2064   // >= max K over all GEMMs (2*EMB = 2048)

__global__ void k_zero(float* __restrict__ z, int n)
{
    int t = blockIdx.x * blockDim.x + threadIdx.x;
    if (t < n) z[t] = 0.f;
}

// ---------------- GEMM fragment helpers -------------------------------------
struct Frag { v2f a, b0, b1, b2, b3; };

__device__ __forceinline__ Frag frag_load(const float* __restrict__ Ar,
                                          const float* __restrict__ B0,
                                          const float* __restrict__ B1,
                                          const float* __restrict__ B2,
                                          const float* __restrict__ B3, int ka)
{
    Frag f;
    f.a.x  = Ar[ka]; f.a.y  = Ar[ka + 1];
    f.b0.x = B0[ka]; f.b0.y = B0[ka + 1];
    f.b1.x = B1[ka]; f.b1.y = B1[ka + 1];
    f.b2.x = B2[ka]; f.b2.y = B2[ka + 1];
    f.b3.x = B3[ka]; f.b3.y = B3[ka + 1];
    return f;
}

__device__ __forceinline__ void frag_mma(v8f& a0, v8f& a1, v8f& a2, v8f& a3,
                                         const Frag& f)
{
    a0 = __builtin_amdgcn_wmma_f32_16x16x4_f32(false, f.a, false, f.b0, (short)0, a0, false, false);
    a1 = __builtin_amdgcn_wmma_f32_16x16x4_f32(false, f.a, false, f.b1, (short)0, a1, false, false);
    a2 = __builtin_amdgcn_wmma_f32_16x16x4_f32(false, f.a, false, f.b2, (short)0, a2, false, false);
    a3 = __builtin_amdgcn_wmma_f32_16x16x4_f32(false, f.a, false, f.b3, (short)0, a3, false, false);
}

// ---------------- GEMM: C = A * B^T, batched --------------------------------
// A: M x K (row stride lda), B: N x K (row stride ldb), C: M x N (ldc).
// One wave computes a 16x64 tile; K stepped by 4 via f32 WMMA.
// tn = ceil(N/64); tiles = ceil(M/16) * tn. zrow: >= K zeros.
__global__ void k_gemm_nt(const float* __restrict__ A, long lda, long sA,
                          const float* __restrict__ Bm, long ldb, long sB,
                          float* __restrict__ C, long ldc, long sC,
                          const float* __restrict__ zrow,
                          int M, int N, int K, int tn, int tiles)
{
    int waveId = threadIdx.x >> 5;
    int lane   = threadIdx.x & 31;
    int tileId = blockIdx.x * (blockDim.x >> 5) + waveId;
    if (tileId >= tiles) return;
    int tM = tileId / tn, tN = tileId % tn;
    const float* Ab = A  + (long)blockIdx.z * sA;
    const float* Bb = Bm + (long)blockIdx.z * sB;
    float*       Cb = C  + (long)blockIdx.z * sC;

    int mrow = lane & 15;
    int kq   = (lane >> 4) << 1;             // 0 or 2: K-pair by half-wave
    int m    = tM * 16 + mrow;
    // OOB rows read a zero-filled row: loop body needs no masking at all.
    const float* Arow = (m < M) ? (Ab + (long)m * lda) : zrow;

    const float* Brow0; const float* Brow1; const float* Brow2; const float* Brow3;
    {
        int n0 = tN * 64 + mrow;
        Brow0 = (n0      < N) ? (Bb + (long)(n0     ) * ldb) : zrow;
        Brow1 = (n0 + 16 < N) ? (Bb + (long)(n0 + 16) * ldb) : zrow;
        Brow2 = (n0 + 32 < N) ? (Bb + (long)(n0 + 32) * ldb) : zrow;
        Brow3 = (n0 + 48 < N) ? (Bb + (long)(n0 + 48) * ldb) : zrow;
    }

    v8f acc0 = {0.f,0.f,0.f,0.f,0.f,0.f,0.f,0.f};
    v8f acc1 = acc0, acc2 = acc0, acc3 = acc0;

    int K4 = K & ~3;       // K-steps of 4
    int KP = K4 & ~7;      // pipelined portion: pairs of K-steps
    if (KP > 0) {
        // ping-pong: two fragment sets, each reloaded right after consumption
        Frag FA = frag_load(Arow, Brow0, Brow1, Brow2, Brow3, kq);
        Frag FB = frag_load(Arow, Brow0, Brow1, Brow2, Brow3, 4 + kq);
        int kk = 0;
        while (kk + 8 < KP) {
            frag_mma(acc0, acc1, acc2, acc3, FA);
            FA = frag_load(Arow, Brow0, Brow1, Brow2, Brow3, kk + 8 + kq);
            frag_mma(acc0, acc1, acc2, acc3, FB);
            FB = frag_load(Arow, Brow0, Brow1, Brow2, Brow3, kk + 12 + kq);
            kk += 8;
        }
        frag_mma(acc0, acc1, acc2, acc3, FA);
        frag_mma(acc0, acc1, acc2, acc3, FB);
    }
    if (K4 > KP) {                           // one leftover 4-step (K4 % 8 == 4)
        Frag F = frag_load(Arow, Brow0, Brow1, Brow2, Brow3, KP + kq);
        frag_mma(acc0, acc1, acc2, acc3, F);
    }
    if (K > K4) {                            // tail (only K%4 != 0, e.g. K=3)
        int ka  = K4 + kq;
        int kc0 = min(ka,     K - 1);
        int kc1 = min(ka + 1, K - 1);
        bool kok0 = (ka     < K), kok1 = (ka + 1 < K);
        float a0 = Arow [kc0], a1 = Arow [kc1];
        float c0 = Brow0[kc0], c1 = Brow0[kc1];
        float d0 = Brow1[kc0], d1 = Brow1[kc1];
        float e0 = Brow2[kc0], e1 = Brow2[kc1];
        float f0 = Brow3[kc0], f1 = Brow3[kc1];
        Frag F;
        F.a.x  = kok0 ? a0 : 0.f; F.a.y  = kok1 ? a1 : 0.f;
        F.b0.x = kok0 ? c0 : 0.f; F.b0.y = kok1 ? c1 : 0.f;
        F.b1.x = kok0 ? d0 : 0.f; F.b1.y = kok1 ? d1 : 0.f;
        F.b2.x = kok0 ? e0 : 0.f; F.b2.y = kok1 ? e1 : 0.f;
        F.b3.x = kok0 ? f0 : 0.f; F.b3.y = kok1 ? f1 : 0.f;
        frag_mma(acc0, acc1, acc2, acc3, F);
    }

    // C/D layout: VGPR r -> row r (lanes 0-15) / row r+8 (lanes 16-31), col = lane&15
    int half = lane >> 4;
    int cn   = tN * 64 + (lane & 15);
#pragma unroll
    for (int r = 0; r < 8; ++r) {
        int cm = tM * 16 + r + 8 * half;
        if (cm < M) {
            long rb = (long)cm * ldc;
            if (cn      < N) Cb[rb + cn     ] = acc0[r];
            if (cn + 16 < N) Cb[rb + cn + 16] = acc1[r];
            if (cn + 32 < N) Cb[rb + cn + 32] = acc2[r];
            if (cn + 48 < N) Cb[rb + cn + 48] = acc3[r];
        }
    }
}

// ---------------- small helpers ---------------------------------------------
__global__ void k_transpose_x(const float* __restrict__ x, float* __restrict__ xt,
                              int B, int C, int N)
{   // x: [B,C,N] -> xt: [B,N,C]
    int t = blockIdx.x * blockDim.x + threadIdx.x;
    int tot = B * C * N;
    if (t >= tot) return;
    int c = t % C, n = (t / C) % N, b = t / (C * N);
    xt[t] = x[((long)b * C + c) * N + n];
}

__global__ void k_make_wcomb(const float* __restrict__ w, float* __restrict__ wc,
                             int Co, int Cin)
{   // w: [Co, 2*Cin] -> wc: [2*Co, Cin] = [wl ; wr - wl]
    int t = blockIdx.x * blockDim.x + threadIdx.x;
    int tot = 2 * Co * Cin;
    if (t >= tot) return;
    int r = t / Cin, c = t - r * Cin;
    if (r < Co) wc[t] = w[(long)r * 2 * Cin + c];
    else {
        int o = r - Co;
        wc[t] = w[(long)o * 2 * Cin + Cin + c] - w[(long)o * 2 * Cin + c];
    }
}

__global__ void k_rownorm(const float* __restrict__ X, long ld, int C,
                          float* __restrict__ xx, int rows)
{
    int r = blockIdx.x * blockDim.x + threadIdx.x;
    if (r >= rows) return;
    float s = 0.f;
    for (int c = 0; c < C; ++c) { float v = X[(long)r * ld + c]; s += v * v; }
    xx[r] = s;
}

// ---------------- top-k (k=20 of N=1024) ------------------------------------
// One wave per row. Each lane caches its 32 column distances in registers;
// 20 rounds of wave argmax (tie -> smallest index, matching jax.lax.top_k).
__global__ void k_topk(const float* __restrict__ P, const float* __restrict__ xx,
                       const uint8_t* __restrict__ mask, int keepLocal,
                       int* __restrict__ idxOut, int N, int Kn)
{
    int lane = threadIdx.x & 31;
    int row  = blockIdx.x * (blockDim.x >> 5) + (threadIdx.x >> 5);
    int b = row / N;
    const float*   prow = P  + (size_t)row * N;
    const float*   xrow = xx + (size_t)b * N;
    const uint8_t* mrow = mask + (size_t)b * N;
    float xn = xx[row];

    float vals[32];                        // N==1024: 32 columns per lane
    unsigned validm = 0u;
#pragma unroll
    for (int t = 0; t < 32; ++t) {
        int j = lane * 32 + t;
        vals[t] = 2.f * prow[j] - xn - xrow[j];   // -||xi - xj||^2
        if ((mrow[j] != 0) == (keepLocal != 0)) validm |= (1u << t);
    }
    unsigned selm = 0u;
    for (int it = 0; it < Kn; ++it) {
        float best = -3.4e38f; int bidx = 1 << 30;
        unsigned avail = validm & ~selm;
#pragma unroll
        for (int t = 0; t < 32; ++t)
            if (avail & (1u << t)) {
                float v = vals[t];
                if (v > best) { best = v; bidx = lane * 32 + t; }  // first hit = low idx
            }
        for (int off = 16; off > 0; off >>= 1) {
            float ov = __shfl_xor(best, off, 32);
            int   oi = __shfl_xor(bidx, off, 32);
            if (ov > best || (ov == best && oi < bidx)) { best = ov; bidx = oi; }
        }
        if (bidx >= (1 << 30)) {
            // fewer kept columns than k: reference fills with row-min entries;
            // pick smallest-index unselected column.
            int cand = 1 << 30;
            unsigned un = ~selm;
#pragma unroll
            for (int t = 0; t < 32; ++t)
                if (un & (1u << t)) { int j = lane * 32 + t; if (j < cand) cand = j; }
            for (int off = 16; off > 0; off >>= 1) {
                int oc = __shfl_xor(cand, off, 32);
                if (oc < cand) cand = oc;
            }
            bidx = cand;
        }
        if ((bidx >> 5) == lane) selm |= 1u << (bidx & 31);
        if (lane == 0) idxOut[(size_t)row * Kn + it] = bidx;
    }
}

// ---------------- BN statistics (chunked, deterministic) --------------------
__global__ void k_edge_stats(const float* __restrict__ U, const int* __restrict__ idx,
                             int Co, int N, int Kn, int total,
                             float* __restrict__ partial, int NCH)
{   // y[e,o] = U[b*N + idx[e], o] + U[bn, Co+o]; partial sums per (channel, chunk)
    __shared__ float s1s[256], s2s[256];
    int o = blockIdx.x, ch = blockIdx.y;
    int per = (total + NCH - 1) / NCH;
    int start = ch * per, end = min(start + per, total);
    long ld = 2L * Co;
    float s1 = 0.f, s2 = 0.f;
    for (int e = start + threadIdx.x; e < end; e += blockDim.x) {
        int bn = e / Kn, b = bn / N;
        int nb = idx[e];
        float y = U[((long)b * N + nb) * ld + o] + U[(long)bn * ld + Co + o];
        s1 += y; s2 += y * y;
    }
    s1s[threadIdx.x] = s1; s2s[threadIdx.x] = s2;
    __syncthreads();
    for (int st = 128; st > 0; st >>= 1) {
        if ((int)threadIdx.x < st) {
            s1s[threadIdx.x] += s1s[threadIdx.x + st];
            s2s[threadIdx.x] += s2s[threadIdx.x + st];
        }
        __syncthreads();
    }
    if (threadIdx.x == 0) {
        partial[((long)o * NCH + ch) * 2 + 0] = s1s[0];
        partial[((long)o * NCH + ch) * 2 + 1] = s2s[0];
    }
}

__global__ void k_col_stats(const float* __restrict__ Y, long ld, int Mrows,
                            float* __restrict__ partial, int NCH)
{
    __shared__ float s1s[256], s2s[256];
    int o = blockIdx.x, ch = blockIdx.y;
    int per = (Mrows + NCH - 1) / NCH;
    int start = ch * per, end = min(start + per, Mrows);
    float s1 = 0.f, s2 = 0.f;
    for (int r = start + threadIdx.x; r < end; r += blockDim.x) {
        float v = Y[(long)r * ld + o];
        s1 += v; s2 += v * v;
    }
    s1s[threadIdx.x] = s1; s2s[threadIdx.x] = s2;
    __syncthreads();
    for (int st = 128; st > 0; st >>= 1) {
        if ((int)threadIdx.x < st) {
            s1s[threadIdx.x] += s1s[threadIdx.x + st];
            s2s[threadIdx.x] += s2s[threadIdx.x + st];
        }
        __syncthreads();
    }
    if (threadIdx.x == 0) {
        partial[((long)o * NCH + ch) * 2 + 0] = s1s[0];
        partial[((long)o * NCH + ch) * 2 + 1] = s2s[0];
    }
}

__global__ void k_stats_final(const float* __restrict__ partial, int NCH, int Cch,
                              float count, float* __restrict__ mean,
                              float* __restrict__ rstd)
{
    int o = blockIdx.x * blockDim.x + threadIdx.x;
    if (o >= Cch) return;
    float s1 = 0.f, s2 = 0.f;
    for (int ch = 0; ch < NCH; ++ch) {
        s1 += partial[((long)o * NCH + ch) * 2 + 0];
        s2 += partial[((long)o * NCH + ch) * 2 + 1];
    }
    float mu  = s1 / count;
    float var = s2 / count - mu * mu;      // biased variance, as jnp.var
    mean[o] = mu;
    rstd[o] = rsqrtf(var + BN_EPS);
}

// ---------------- edge epilogue: gather + BN + LReLU + max over k -----------
__global__ void k_edge_apply(const float* __restrict__ U, const int* __restrict__ idx,
                             const float* __restrict__ mean, const float* __restrict__ rstd,
                             const float* __restrict__ g, const float* __restrict__ bt,
                             float* __restrict__ out, int ldo, int outOff,
                             int Co, int N, int Kn, int total)
{
    int t = blockIdx.x * blockDim.x + threadIdx.x;
    if (t >= total) return;
    int bn = t / Co, o = t - bn * Co;
    int b = bn / N;
    long ld = 2L * Co;
    float z  = U[(long)bn * ld + Co + o];
    float mu = mean[o], rs = rstd[o], gg = g[o], bb = bt[o];
    const int* ip = idx + (long)bn * Kn;
    float mx = -3.4e38f;
    for (int kk = 0; kk < Kn; ++kk) {
        int nb = ip[kk];
        float y = U[((long)b * N + nb) * ld + o] + z;
        y = (y - mu) * rs * gg + bb;
        y = LRELU(y);
        mx = fmaxf(mx, y);
    }
    out[(long)bn * ldo + outOff + o] = mx;
}

// ---------------- misc elementwise ------------------------------------------
__global__ void k_merge_xm(const float* __restrict__ cL, const float* __restrict__ cG,
                           const uint8_t* __restrict__ mask, float* __restrict__ xm,
                           int Cw, int total)
{
    int t = blockIdx.x * blockDim.x + threadIdx.x;
    if (t >= total) return;
    int row = t / Cw;
    xm[t] = mask[row] ? cL[t] : cG[t];
}

__global__ void k_bn_apply(float* __restrict__ Y, const float* __restrict__ mean,
                           const float* __restrict__ rstd, const float* __restrict__ g,
                           const float* __restrict__ bt, int Cch, int total)
{
    int t = blockIdx.x * blockDim.x + threadIdx.x;
    if (t >= total) return;
    int o = t % Cch;
    float v = (Y[t] - mean[o]) * rstd[o] * g[o] + bt[o];
    Y[t] = LRELU(v);
}

__global__ void k_pool(const float* __restrict__ Y, float* __restrict__ p,
                       int N, int Cch)
{   // one wave per (b, channel): max + mean over N points
    int lane = threadIdx.x & 31;
    int id = blockIdx.x * (blockDim.x >> 5) + (threadIdx.x >> 5);
    int b = id / Cch, c = id - b * Cch;
    float mx = -3.4e38f, sm = 0.f;
    for (int n = lane; n < N; n += 32) {
        float v = Y[((long)b * N + n) * Cch + c];
        mx = fmaxf(mx, v); sm += v;
    }
    for (int off = 16; off > 0; off >>= 1) {
        mx = fmaxf(mx, __shfl_xor(mx, off, 32));
        sm += __shfl_xor(sm, off, 32);
    }
    if (lane == 0) {
        p[(long)b * 2 * Cch + c]        = mx;
        p[(long)b * 2 * Cch + Cch + c]  = sm / (float)N;
    }
}

__global__ void k_bn_rows(const float* __restrict__ Yin, const float* __restrict__ bias,
                          const float* __restrict__ g, const float* __restrict__ bt,
                          float* __restrict__ Yout, int Bsz, int Cch)
{   // BN over the batch dim (8 samples) + optional bias + LReLU; 1 thread/channel
    int c = blockIdx.x * blockDim.x + threadIdx.x;
    if (c >= Cch) return;
    float bv = bias ? bias[c] : 0.f;
    float s = 0.f, s2 = 0.f;
    for (int b = 0; b < Bsz; ++b) {
        float v = Yin[(long)b * Cch + c] + bv;
        s += v; s2 += v * v;
    }
    float mu = s / (float)Bsz;
    float var = s2 / (float)Bsz - mu * mu;
    float rs = rsqrtf(var + BN_EPS);
    float gg = g[c], bb = bt[c];
    for (int b = 0; b < Bsz; ++b) {
        float v = Yin[(long)b * Cch + c] + bv;
        float y = (v - mu) * rs * gg + bb;
        Yout[(long)b * Cch + c] = LRELU(y);
    }
}

__global__ void k_bias_add(float* __restrict__ out, const float* __restrict__ bias,
                           int Cch, int total)
{
    int t = blockIdx.x * blockDim.x + threadIdx.x;
    if (t >= total) return;
    out[t] += bias[t % Cch];   // `out` was fully overwritten by the GEMM this call
}

// ---------------- host orchestration ----------------------------------------
static inline void launch_gemm_nt(hipStream_t s, const float* A, long lda, long sA,
                                  const float* B, long ldb, long sB,
                                  float* C, long ldc, long sC,
                                  const float* zrow,
                                  int M, int N, int K, int batch)
{
    int tm = (M + 15) >> 4, tn = (N + 63) >> 6;   // 16 x 64 wave tiles
    int tiles = tm * tn;
    dim3 grid((tiles + 3) / 4, 1, batch);
    hipLaunchKernelGGL(k_gemm_nt, grid, dim3(128), 0, s,
                       A, lda, sA, B, ldb, sB, C, ldc, sC, zrow, M, N, K, tn, tiles);
}

extern "C" void kernel_launch(void* const* d_in, const int* in_sizes, int n_in,
                              void* d_out, int out_size, void* d_ws, size_t ws_size,
                              hipStream_t stream)
{
    (void)in_sizes; (void)n_in; (void)out_size; (void)ws_size;
    const int B = 8, N = 1024, Kn = 20, NCH = 32, CW = 512, EMB = 1024;
    const int BN = B * N;

    // inputs, in setup_inputs() dict-insertion order
    const float*   x    = (const float*)d_in[0];
    const uint8_t* mask = (const uint8_t*)d_in[1];   // jax bool_ = 1 byte/elem
    const float* w[9];
    for (int i = 0; i < 9; ++i) w[i] = (const float*)d_in[2 + i];
    const float* l1w = (const float*)d_in[11];
    const float* g10 = (const float*)d_in[12];
    const float* b10 = (const float*)d_in[13];
    const float* l2w = (const float*)d_in[14];
    const float* l2b = (const float*)d_in[15];
    const float* g11 = (const float*)d_in[16];
    const float* b11 = (const float*)d_in[17];
    const float* l3w = (const float*)d_in[18];
    const float* l3b = (const float*)d_in[19];
    const float *gv[9], *bv[9];
    for (int j = 0; j < 9; ++j) {
        gv[j] = (const float*)d_in[20 + 2 * j];
        bv[j] = (const float*)d_in[21 + 2 * j];
    }
    float* out = (float*)d_out;

    // workspace bump allocator (~102 MB total)
    char* wsb = (char*)d_ws;
    size_t off = 0;
    auto alloc = [&](size_t bytes) -> void* {
        off = (off + 255) & ~(size_t)255;
        void* p = wsb + off;
        off += bytes;
        return p;
    };
    float* xt    = (float*)alloc((size_t)BN * 3 * 4);
    float* cL    = (float*)alloc((size_t)BN * CW * 4);
    float* cG    = (float*)alloc((size_t)BN * CW * 4);
    float* U     = (float*)alloc((size_t)BN * 512 * 4);        // 2*Co_max = 512
    float* xm    = (float*)alloc((size_t)BN * CW * 4);
    float* big   = (float*)alloc((size_t)B * N * N * 4);       // P, reused as y9
    float* xx    = (float*)alloc((size_t)BN * 4);
    int*   idx   = (int*)  alloc((size_t)BN * Kn * 4);
    float* wcomb = (float*)alloc((size_t)512 * 128 * 4);
    float* part  = (float*)alloc((size_t)EMB * NCH * 2 * 4);
    float* meanb = (float*)alloc((size_t)EMB * 4);
    float* rstdb = (float*)alloc((size_t)EMB * 4);
    float* pbuf  = (float*)alloc((size_t)B * 2 * EMB * 4);
    float* h1    = (float*)alloc((size_t)B * 512 * 4);
    float* h1n   = (float*)alloc((size_t)B * 512 * 4);
    float* h2    = (float*)alloc((size_t)B * 256 * 4);
    float* h2n   = (float*)alloc((size_t)B * 256 * 4);
    float* zrow  = (float*)alloc((size_t)ZLEN * 4);            // zero row for OOB tiles

    // init zero row (ws is poisoned; must be rewritten every call)
    hipLaunchKernelGGL(k_zero, dim3((ZLEN + 255) / 256), dim3(256), 0, stream, zrow, ZLEN);

    // x [B,3,N] -> xt [B,N,3]
    {
        int tot = BN * 3;
        hipLaunchKernelGGL(k_transpose_x, dim3((tot + 255) / 256), dim3(256), 0, stream,
                           x, xt, B, 3, N);
    }

    // one edge-conv block
    auto edge_block = [&](const float* Xin, int ldin, int Cin, int Co,
                          const float* wB, const float* gB, const float* btB,
                          int keepLocal, float* outBase, int outOff) {
        int nw = 2 * Co * Cin;
        hipLaunchKernelGGL(k_make_wcomb, dim3((nw + 255) / 256), dim3(256), 0, stream,
                           wB, wcomb, Co, Cin);
        // U = Xin * [wl ; wr-wl]^T : [8192 x 2Co]
        launch_gemm_nt(stream, Xin, ldin, 0, wcomb, Cin, 0,
                       U, 2 * Co, 0, zrow, BN, 2 * Co, Cin, 1);
        // P[b] = Xin_b * Xin_b^T : [1024 x 1024] per batch
        launch_gemm_nt(stream, Xin, ldin, (long)N * ldin, Xin, ldin, (long)N * ldin,
                       big, N, (long)N * N, zrow, N, N, Cin, B);
        hipLaunchKernelGGL(k_rownorm, dim3((BN + 255) / 256), dim3(256), 0, stream,
                           Xin, (long)ldin, Cin, xx, BN);
        hipLaunchKernelGGL(k_topk, dim3(BN / 4), dim3(128), 0, stream,
                           big, xx, mask, keepLocal, idx, N, Kn);
        int totalE = BN * Kn;
        hipLaunchKernelGGL(k_edge_stats, dim3(Co, NCH), dim3(256), 0, stream,
                           U, idx, Co, N, Kn, totalE, part, NCH);
        hipLaunchKernelGGL(k_stats_final, dim3((Co + 255) / 256), dim3(256), 0, stream,
                           part, NCH, Co, (float)totalE, meanb, rstdb);
        int totA = BN * Co;
        hipLaunchKernelGGL(k_edge_apply, dim3((totA + 255) / 256), dim3(256), 0, stream,
                           U, idx, meanb, rstdb, gB, btB,
                           outBase, CW, outOff, Co, N, Kn, totA);
    };

    // local branch (keep = mask), outputs packed into cL at channel offsets
    edge_block(xt,       3,   3,   64,  w[0], gv[0], bv[0], 1, cL, 0);
    edge_block(cL + 0,   CW,  64,  64,  w[1], gv[1], bv[1], 1, cL, 64);
    edge_block(cL + 64,  CW,  64,  128, w[2], gv[2], bv[2], 1, cL, 128);
    edge_block(cL + 128, CW,  128, 256, w[3], gv[3], bv[3], 1, cL, 256);
    // global branch (keep = !mask)
    edge_block(xt,       3,   3,   64,  w[4], gv[4], bv[4], 0, cG, 0);
    edge_block(cG + 0,   CW,  64,  64,  w[5], gv[5], bv[5], 0, cG, 64);
    edge_block(cG + 64,  CW,  64,  128, w[6], gv[6], bv[6], 0, cG, 128);
    edge_block(cG + 128, CW,  128, 256, w[7], gv[7], bv[7], 0, cG, 256);

    // scatter-merge by mask
    {
        int tot = BN * CW;
        hipLaunchKernelGGL(k_merge_xm, dim3((tot + 255) / 256), dim3(256), 0, stream,
                           cL, cG, mask, xm, CW, tot);
    }

    // conv9: y9 = xm * w9^T [8192 x 1024], BN over (B,N), LReLU (in place)
    float* y9 = big;
    launch_gemm_nt(stream, xm, CW, 0, w[8], CW, 0, y9, EMB, 0, zrow, BN, EMB, CW, 1);
    hipLaunchKernelGGL(k_col_stats, dim3(EMB, NCH), dim3(256), 0, stream,
                       y9, (long)EMB, BN, part, NCH);
    hipLaunchKernelGGL(k_stats_final, dim3((EMB + 255) / 256), dim3(256), 0, stream,
                       part, NCH, EMB, (float)BN, meanb, rstdb);
    {
        int tot = BN * EMB;
        hipLaunchKernelGGL(k_bn_apply, dim3((tot + 255) / 256), dim3(256), 0, stream,
                           y9, meanb, rstdb, gv[8], bv[8], EMB, tot);
    }
    // global max+mean pooling -> p [B, 2*EMB]
    hipLaunchKernelGGL(k_pool, dim3(B * EMB / 4), dim3(128), 0, stream, y9, pbuf, N, EMB);

    // MLP head
    launch_gemm_nt(stream, pbuf, 2 * EMB, 0, l1w, 2 * EMB, 0, h1, 512, 0, zrow, B, 512, 2 * EMB, 1);
    hipLaunchKernelGGL(k_bn_rows, dim3(2), dim3(256), 0, stream,
                       h1, (const float*)nullptr, g10, b10, h1n, B, 512);
    launch_gemm_nt(stream, h1n, 512, 0, l2w, 512, 0, h2, 256, 0, zrow, B, 256, 512, 1);
    hipLaunchKernelGGL(k_bn_rows, dim3(1), dim3(256), 0, stream,
                       h2, l2b, g11, b11, h2n, B, 256);
    launch_gemm_nt(stream, h2n, 256, 0, l3w, 256, 0, out, 40, 0, zrow, B, 40, 256, 1);
    hipLaunchKernelGGL(k_bias_add, dim3(2), dim3(256), 0, stream, out, l3b, 40, B * 40);
}